// GraphConvolution_27573690040485
// MI455X (gfx1250) — compile-verified
//
#include <hip/hip_runtime.h>
#include <hip/hip_bf16.h>
#include <stdint.h>

// Problem constants (match reference)
#define NB    8
#define NN    4096
#define FIN   256
#define FOUT  256
#define NE    131072

typedef __attribute__((ext_vector_type(2))) float        v2f;
typedef __attribute__((ext_vector_type(8))) float        v8f;
typedef __attribute__((ext_vector_type(4))) unsigned int v4u;
typedef __attribute__((ext_vector_type(8))) int          v8i;
typedef __attribute__((ext_vector_type(4))) int          v4i;

#if defined(__HIP_DEVICE_COMPILE__)
#if !__has_builtin(__builtin_amdgcn_tensor_load_to_lds)
#error "tensor_load_to_lds builtin not available on this device toolchain"
#endif
#endif

// ---------------------------------------------------------------------------
// Issue one 2-D TDM load (global -> LDS), per CDNA5 ISA D# layout (§8.3-8.5).
//   tile_x  : tile extent along dim0 (contiguous floats)
//   tile_y  : tile extent along dim1 (rows)
//   tdim0/1 : tensor extents (data_size units)
//   stride0 : row stride of the tensor (data_size units)
// data_size = 4 bytes (encoded 2). Groups 2/3 zero-filled (2-D tensor).
__device__ __forceinline__ void tdm_load_2d(uint64_t gaddr, uint32_t lds_addr,
                                            uint32_t tile_x, uint32_t tile_y,
                                            uint32_t tdim0, uint32_t tdim1,
                                            uint32_t stride0) {
#if defined(__HIP_DEVICE_COMPILE__)
    v4u g0;
    g0.x = 1u;                                     // count=1, user descriptor
    g0.y = lds_addr;                               // lds_addr [63:32]
    g0.z = (uint32_t)gaddr;                        // global_addr lo [95:64]
    g0.w = (uint32_t)(gaddr >> 32) | (2u << 30);   // global_addr hi, type=2
    v8i g1;
    g1[0] = (int)(2u << 16);                       // data_size = 4B
    g1[1] = (int)(tdim0 << 16);                    // tensor_dim0 lo16 @ [63:48]
    g1[2] = (int)((tdim0 >> 16) | (tdim1 << 16));  // dim0 hi / dim1 lo
    g1[3] = (int)((tdim1 >> 16) | (tile_x << 16)); // dim1 hi / tile_dim0
    g1[4] = (int)(tile_y & 0xffffu);               // tile_dim1 (tile_dim2=0)
    g1[5] = (int)stride0;                          // tensor_dim0_stride lo32
    g1[6] = 0;
    g1[7] = 0;
    v4i g2 = {0, 0, 0, 0};
    v4i g3 = {0, 0, 0, 0};
#if __clang_major__ >= 23
    v8i g4 = {0, 0, 0, 0, 0, 0, 0, 0};
    __builtin_amdgcn_tensor_load_to_lds(g0, g1, g2, g3, g4, 0);
#else
    __builtin_amdgcn_tensor_load_to_lds(g0, g1, g2, g3, 0);
#endif
#else
    (void)gaddr; (void)lds_addr; (void)tile_x; (void)tile_y;
    (void)tdim0; (void)tdim1; (void)stride0;
#endif
}

__device__ __forceinline__ void gc_wait_tensorcnt0() {
#if defined(__HIP_DEVICE_COMPILE__)
    __builtin_amdgcn_s_wait_tensorcnt(0);
#endif
}

// ---------------------------------------------------------------------------
__global__ void gc_zero_u32(unsigned int* __restrict__ p, int n) {
    int i = blockIdx.x * blockDim.x + threadIdx.x;
    if (i < n) p[i] = 0u;
}

// Dedupe edges via dense bitmap; count distinct out-degree per source row.
__global__ void gc_build_adj(const int* __restrict__ ei,
                             unsigned int* __restrict__ bitmap,
                             int* __restrict__ degi,
                             unsigned char* __restrict__ flag) {
    int e = blockIdx.x * blockDim.x + threadIdx.x;
    if (e >= NE) return;
    int s = ei[e];
    int d = ei[NE + e];
    unsigned int pos = (unsigned int)s * (unsigned int)NN + (unsigned int)d;
    unsigned int bit = 1u << (pos & 31u);
    unsigned int old = atomicOr(&bitmap[pos >> 5], bit);
    if (!(old & bit)) {
        atomicAdd(&degi[s], 1);
        flag[e] = 1;
    }
}

__global__ void gc_inv_deg(const int* __restrict__ degi,
                           float* __restrict__ invdeg) {
    int i = blockIdx.x * blockDim.x + threadIdx.x;
    if (i < NN) invdeg[i] = 1.0f / ((float)degi[i] + 1e-6f);
}

// Sparse aggregation for one batch: agg[s,:] += x_b[d,:] over distinct edges.
__global__ void gc_scatter(const int* __restrict__ ei,
                           const unsigned char* __restrict__ flag,
                           const float* __restrict__ xb,
                           float* __restrict__ agg) {
    int e = blockIdx.x * 4 + (threadIdx.x >> 6);
    int t = threadIdx.x & 63;
    if (!flag[e]) return;
    int s = ei[e];
    int d = ei[NE + e];
    const float4 v = reinterpret_cast<const float4*>(xb + (size_t)d * FIN)[t];
    float* dst = agg + (size_t)s * FIN + t * 4;
    atomicAdd(dst + 0, v.x);
    atomicAdd(dst + 1, v.y);
    atomicAdd(dst + 2, v.z);
    atomicAdd(dst + 3, v.w);
}

// ---------------------------------------------------------------------------
// out_b = diag(invdeg) * (agg @ W) + bias.
// 512 threads = 16 waves; block computes a 64x64 macro-tile (4x4 wave tiles).
// K consumed in two 128-chunks; each chunk is staged into LDS by the Tensor
// Data Mover (wave 0 issues 2 descriptors, waits on TENSORcnt), then all
// waves run V_WMMA_F32_16X16X4_F32 out of LDS.
//   A slab : 64 rows x 128 k  (32 KB)   sA[r*128 + k]
//   W panel: 128 k  x 64 cols (32 KB)   sW[k*64  + c]
__global__ void gc_gemm_wmma(const float* __restrict__ agg,
                             const float* __restrict__ W,
                             const float* __restrict__ invdeg,
                             const float* __restrict__ bias,
                             float* __restrict__ outb) {
    __shared__ float sA[64 * 128];
    __shared__ float sW[128 * 64];

    const int lane = threadIdx.x & 31;
    const int wave = threadIdx.x >> 5;
    const int mt   = wave >> 2;              // 0..3 : M sub-tile
    const int nt   = wave & 3;               // 0..3 : N sub-tile
    const int half = lane >> 4;              // K half-pair select
    const int mr   = lane & 15;
    const int blockRow0 = blockIdx.x * 64;
    const int blockCol0 = blockIdx.y * 64;

    v8f c = {};
    for (int kc = 0; kc < FIN; kc += 128) {
        if (threadIdx.x < 32) {              // wave 0 drives the TDM
            uint64_t ga = (uint64_t)(uintptr_t)(agg + (size_t)blockRow0 * FIN + kc);
            uint64_t gw = (uint64_t)(uintptr_t)(W + (size_t)kc * FOUT + blockCol0);
            tdm_load_2d(ga, (uint32_t)(uintptr_t)&sA[0], 128u, 64u,
                        (uint32_t)FIN, (uint32_t)NN, (uint32_t)FIN);
            tdm_load_2d(gw, (uint32_t)(uintptr_t)&sW[0], 64u, 128u,
                        (uint32_t)FOUT, (uint32_t)FIN, (uint32_t)FOUT);
        gc_wait_tensorcnt0();
        }
        __syncthreads();                     // LDS tiles visible to all waves

        const float* aRow  = &sA[(mt * 16 + mr) * 128];
        const float* wBase = &sW[nt * 16 + mr];
#pragma unroll 8
        for (int k0 = 0; k0 < 128; k0 += 4) {
            const int ka = k0 + 2 * half;
            v2f a, b;
            a.x = aRow[ka];
            a.y = aRow[ka + 1];
            b.x = wBase[(size_t)ka * 64];
            b.y = wBase[(size_t)(ka + 1) * 64];
            c = __builtin_amdgcn_wmma_f32_16x16x4_f32(
                    false, a, false, b, (short)0, c, false, false);
        }
        __syncthreads();                     // tiles consumed; safe to refill
    }

    const int row0 = blockRow0 + mt * 16;
    const int col  = blockCol0 + nt * 16 + mr;
    const float bv = bias[col];
#pragma unroll
    for (int m = 0; m < 8; ++m) {
        const int row = row0 + m + 8 * half;
        outb[(size_t)row * FOUT + col] = c[m] * invdeg[row] + bv;
    }
}

// ---------------------------------------------------------------------------
extern "C" void kernel_launch(void* const* d_in, const int* in_sizes, int n_in,
                              void* d_out, int out_size, void* d_ws, size_t ws_size,
                              hipStream_t stream) {
    const float* x    = (const float*)d_in[0];   // (B, N, FIN)
    const int*   ei   = (const int*)  d_in[1];   // (2, E)
    const float* W    = (const float*)d_in[2];   // (FIN, FOUT)
    const float* bias = (const float*)d_in[3];   // (FOUT,)
    float*       out  = (float*)d_out;           // (B, N, FOUT)

    // Workspace layout
    char* ws = (char*)d_ws;
    const size_t BITMAP_B = (size_t)NN * NN / 8;          // 2 MB
    unsigned int*  bitmap = (unsigned int*)ws;
    int*           degi   = (int*)(ws + BITMAP_B);
    float*         invdeg = (float*)(ws + BITMAP_B + (size_t)NN * 4);
    unsigned char* flag   = (unsigned char*)(ws + BITMAP_B + (size_t)NN * 8);
    float*         agg    = (float*)(ws + BITMAP_B + (size_t)NN * 8 + NE);

    // Zero bitmap + degi + invdeg + flag in one sweep (contiguous region)
    {
        int nwords = (int)((BITMAP_B + (size_t)NN * 8 + NE) / 4);
        gc_zero_u32<<<(nwords + 255) / 256, 256, 0, stream>>>(bitmap, nwords);
    }

    // Build dedup'd adjacency + degrees (once; shared across batches)
    gc_build_adj<<<NE / 256, 256, 0, stream>>>(ei, bitmap, degi, flag);
    gc_inv_deg<<<NN / 256, 256, 0, stream>>>(degi, invdeg);

    // Per batch: zero agg -> sparse aggregate -> TDM-staged WMMA GEMM
    const int aggWords = NN * FIN;
    for (int b = 0; b < NB; ++b) {
        gc_zero_u32<<<(aggWords + 255) / 256, 256, 0, stream>>>(
            (unsigned int*)agg, aggWords);
        gc_scatter<<<NE / 4, 256, 0, stream>>>(
            ei, flag, x + (size_t)b * NN * FIN, agg);
        dim3 grid(NN / 64, FOUT / 64);
        gc_gemm_wmma<<<grid, 512, 0, stream>>>(
            agg, W, invdeg, bias, out + (size_t)b * NN * FOUT);
    }
}